// E8P12Codebook_33449205301885
// MI455X (gfx1250) — compile-verified
//
#include <hip/hip_runtime.h>
#include <math.h>
#include <stdint.h>

// CDNA5 / gfx1250: wave32, WMMA f32 16x16x4 (two K=4 steps == exact K=8 fp32 dot),
// codebook staged into LDS with GLOBAL_LOAD_ASYNC_TO_LDS (ASYNCcnt).
typedef float v2f __attribute__((ext_vector_type(2)));
typedef float v8f __attribute__((ext_vector_type(8)));

#define WAVES_PER_BLOCK 8
#define ROWS_PER_WAVE   16
#define ROWS_PER_BLOCK  (WAVES_PER_BLOCK * ROWS_PER_WAVE) // 128
#define GMAX            1536  // LDS-staged codebook capacity (49KB grid + 6KB norms)

// Async copy: one dword / four dwords from global (SADDR base + VGPR byte offset)
// into LDS at per-lane byte offset. Tracked by ASYNCcnt.
__device__ __forceinline__ void async_b32_to_lds(uint32_t lds_off, const void* gbase,
                                                 uint32_t voff_bytes) {
    asm volatile("global_load_async_to_lds_b32 %0, %1, %2"
                 :
                 : "v"(lds_off), "v"(voff_bytes), "s"(gbase)
                 : "memory");
}
__device__ __forceinline__ void async_b128_to_lds(uint32_t lds_off, const void* gbase,
                                                  uint32_t voff_bytes) {
    asm volatile("global_load_async_to_lds_b128 %0, %1, %2"
                 :
                 : "v"(lds_off), "v"(voff_bytes), "s"(gbase)
                 : "memory");
}
__device__ __forceinline__ void wait_asynccnt0() {
    asm volatile("s_wait_asynccnt 0x0" ::: "memory");
}

__device__ __forceinline__ uint32_t lds_offset_u32(const void* shared_ptr) {
    // Flat address of a __shared__ object: aperture in [63:32], LDS offset in [31:0].
    return (uint32_t)(uintptr_t)shared_ptr;
}

__device__ __forceinline__ void e8p_finalize_variant(
    const float xv[8], float shift, int parity, int ridx,
    const float* __restrict__ grid,
    const int* __restrict__ part_abs_map,
    const unsigned char* __restrict__ grid_abs_odd,
    float vals[8], int* code, float* err2)
{
    float xc[8];
    int cnt = 0;
#pragma unroll
    for (int i = 0; i < 8; ++i) { xc[i] = xv[i] + shift; cnt += (xc[i] < 0.f) ? 1 : 0; }
    const float flip = (cnt & 1) ? -1.f : 1.f;
    float mask[8];
#pragma unroll
    for (int i = 0; i < 8; ++i) mask[i] = (xc[i] < 0.f) ? -1.f : 1.f;
    mask[7] *= flip;

    const float* gr = grid + (size_t)ridx * 8;
    float rounded[8];
#pragma unroll
    for (int i = 0; i < 8; ++i) rounded[i] = gr[i];

    float e2 = 0.f;
#pragma unroll
    for (int i = 0; i < 8; ++i) {
        vals[i] = rounded[i] * mask[i];
        const float d = xc[i] - vals[i];
        e2 += d * d;
    }

    const int abs_idx = part_abs_map[ridx];
    int pre[8];
#pragma unroll
    for (int i = 0; i < 8; ++i) pre[i] = ((rounded[i] < 0.f) != (mask[i] < 0.f)) ? 1 : 0;
    const int SP[8] = {0, 2, 4, 6, 1, 3, 5, 7}; // _SIGN_PERM
    int sm[8];
#pragma unroll
    for (int j = 0; j < 8; ++j) sm[j] = pre[SP[j]];
    sm[7] ^= (grid_abs_odd[abs_idx] ? 1 : 0);
    sm[0] ^= parity;

    int c = abs_idx << 8;
#pragma unroll
    for (int j = 0; j < 8; ++j) c += sm[j] << j;
    *code = c;
    *err2 = e2;
}

__global__ __launch_bounds__(256) void e8p12_quantize_wmma_kernel(
    const float* __restrict__ x,
    const float* __restrict__ grid,
    const float* __restrict__ gnorm,
    const int* __restrict__ part_abs_map,
    const unsigned char* __restrict__ grid_abs_odd,
    float* __restrict__ out_vals,
    int* __restrict__ out_codes,
    int N, int G)
{
    __shared__ float sGrid[GMAX * 8];
    __shared__ float sNorm[GMAX];
    __shared__ int   sIdxP[WAVES_PER_BLOCK][ROWS_PER_WAVE];
    __shared__ int   sIdxM[WAVES_PER_BLOCK][ROWS_PER_WAVE];

    const int tid  = threadIdx.x;
    const int wave = tid >> 5;
    const int lane = tid & 31;
    const int n    = lane & 15;   // column / row-in-tile selector
    const int hi   = lane >> 4;   // K-half (A/B) or M-half (C/D)
    const int blockBase = blockIdx.x * ROWS_PER_BLOCK;
    const int waveBase  = blockBase + wave * ROWS_PER_WAVE;
    const bool useLds   = (G <= GMAX);

    // ---- Stage codebook + norms into LDS with async-to-LDS (ASYNCcnt) ----
    if (useLds) {
        const uint32_t gBase = lds_offset_u32(sGrid);
        const uint32_t nBase = lds_offset_u32(sNorm);
        const int nVec = G * 2;              // G*8 floats as float4s
        for (int i = tid; i < nVec; i += 256)
            async_b128_to_lds(gBase + (uint32_t)i * 16u, grid, (uint32_t)i * 16u);
        for (int i = tid; i < G; i += 256)
            async_b32_to_lds(nBase + (uint32_t)i * 4u, gnorm, (uint32_t)i * 4u);
        wait_asynccnt0();
    }
    __syncthreads();

    // ---------------- Phase 1: WMMA scoring + argmax over grid ----------------
    {
        int r  = waveBase + n;
        int rc = (r < N) ? r : (N - 1); // clamp; tail rows compute unused junk
        const float* xr = x + (size_t)rc * 8;
        float xv[8];
#pragma unroll
        for (int i = 0; i < 8; ++i) xv[i] = xr[i];

        // x_part for plus (x+0.25) and minus (x-0.25) variants
        float xpP[8], xpM[8];
        {
            float xc[8]; int cnt = 0;
#pragma unroll
            for (int i = 0; i < 8; ++i) { xc[i] = xv[i] + 0.25f; cnt += (xc[i] < 0.f) ? 1 : 0; }
            const float flip = (cnt & 1) ? -1.f : 1.f;
#pragma unroll
            for (int i = 0; i < 8; ++i) xpP[i] = fabsf(xc[i]);
            xpP[7] *= flip;
        }
        {
            float xc[8]; int cnt = 0;
#pragma unroll
            for (int i = 0; i < 8; ++i) { xc[i] = xv[i] - 0.25f; cnt += (xc[i] < 0.f) ? 1 : 0; }
            const float flip = (cnt & 1) ? -1.f : 1.f;
#pragma unroll
            for (int i = 0; i < 8; ++i) xpM[i] = fabsf(xc[i]);
            xpM[7] *= flip;
        }

        // A fragments (16x4 f32 layout): K = vgpr + 2*hi (+4 for step 1).
        v2f aP0, aP1, aM0, aM1;
        aP0[0] = xpP[2 * hi + 0]; aP0[1] = xpP[2 * hi + 1];
        aP1[0] = xpP[2 * hi + 4]; aP1[1] = xpP[2 * hi + 5];
        aM0[0] = xpM[2 * hi + 0]; aM0[1] = xpM[2 * hi + 1];
        aM1[0] = xpM[2 * hi + 4]; aM1[1] = xpM[2 * hi + 5];

        float bestP[8], bestM[8];
        int   idxP[8],  idxM[8];
#pragma unroll
        for (int v = 0; v < 8; ++v) {
            bestP[v] = -INFINITY; bestM[v] = -INFINITY;
            idxP[v] = 0; idxM[v] = 0;
        }

#pragma unroll 2
        for (int g0 = 0; g0 < G; g0 += 16) {
            const int gi = g0 + n;

            // B fragment (4x16 f32): N = lane&15, K = vgpr + 2*hi (+4 step 1).
            v2f b0, b1;
            float ci; // C init: -0.5*||g||^2 -> WMMA computes dot - 0.5*nrm directly
            if (gi < G) {
                if (useLds) {
                    const int gb = gi * 8;
                    b0[0] = sGrid[gb + 2 * hi + 0]; b0[1] = sGrid[gb + 2 * hi + 1];
                    b1[0] = sGrid[gb + 2 * hi + 4]; b1[1] = sGrid[gb + 2 * hi + 5];
                    ci = -0.5f * sNorm[gi];
                } else {
                    __builtin_prefetch(grid + (size_t)(g0 + 16) * 8, 0, 0);
                    const float* gr = grid + (size_t)gi * 8;
                    b0[0] = gr[2 * hi + 0]; b0[1] = gr[2 * hi + 1];
                    b1[0] = gr[2 * hi + 4]; b1[1] = gr[2 * hi + 5];
                    ci = -0.5f * gnorm[gi];
                }
            } else {
                b0[0] = 0.f; b0[1] = 0.f; b1[0] = 0.f; b1[1] = 0.f;
                ci = -INFINITY; // pad columns never win the argmax
            }

            v8f cP = {ci, ci, ci, ci, ci, ci, ci, ci};
            v8f cM = {ci, ci, ci, ci, ci, ci, ci, ci};
            // D = A*B + C, fp32, K=4 per op; chain two for K=8.
            cP = __builtin_amdgcn_wmma_f32_16x16x4_f32(false, aP0, false, b0, (short)0, cP, false, false);
            cP = __builtin_amdgcn_wmma_f32_16x16x4_f32(false, aP1, false, b1, (short)0, cP, false, false);
            cM = __builtin_amdgcn_wmma_f32_16x16x4_f32(false, aM0, false, b0, (short)0, cM, false, false);
            cM = __builtin_amdgcn_wmma_f32_16x16x4_f32(false, aM1, false, b1, (short)0, cM, false, false);

            // C/D layout: VGPR v, lanes 0-15 -> M=v, lanes 16-31 -> M=v+8; N = lane&15.
            // argmax(2c) == argmax(c): compare accumulators directly.
#pragma unroll
            for (int v = 0; v < 8; ++v) {
                if (cP[v] > bestP[v]) { bestP[v] = cP[v]; idxP[v] = gi; }
                if (cM[v] > bestM[v]) { bestM[v] = cM[v]; idxM[v] = gi; }
            }
        }

        // Argmax reduce across the 16 lanes holding each row (xor stays in half).
#pragma unroll
        for (int v = 0; v < 8; ++v) {
#pragma unroll
            for (int s = 8; s >= 1; s >>= 1) {
                float ob = __shfl_xor(bestP[v], s, 32);
                int   oi = __shfl_xor(idxP[v],  s, 32);
                if (ob > bestP[v] || (ob == bestP[v] && oi < idxP[v])) { bestP[v] = ob; idxP[v] = oi; }
                ob = __shfl_xor(bestM[v], s, 32);
                oi = __shfl_xor(idxM[v],  s, 32);
                if (ob > bestM[v] || (ob == bestM[v] && oi < idxM[v])) { bestM[v] = ob; idxM[v] = oi; }
            }
        }

        if (n == 0) { // lane 0 (rows 0..7) and lane 16 (rows 8..15) publish winners
#pragma unroll
            for (int v = 0; v < 8; ++v) {
                sIdxP[wave][v + 8 * hi] = idxP[v];
                sIdxM[wave][v + 8 * hi] = idxM[v];
            }
        }
    }
    __syncthreads();

    // ---------------- Phase 2: per-row reconstruction + selection ----------------
    if (tid < ROWS_PER_BLOCK) {
        const int r = blockBase + tid;
        if (r < N) {
            const int ip = sIdxP[tid >> 4][tid & 15];
            const int im = sIdxM[tid >> 4][tid & 15];
            const float* xr = x + (size_t)r * 8;
            float xv[8];
#pragma unroll
            for (int i = 0; i < 8; ++i) xv[i] = xr[i];

            float valsP[8], valsM[8];
            int codeP, codeM;
            float e2P, e2M;
            e8p_finalize_variant(xv, +0.25f, 1, ip, grid, part_abs_map, grid_abs_odd,
                                 valsP, &codeP, &e2P);
            e8p_finalize_variant(xv, -0.25f, 0, im, grid, part_abs_map, grid_abs_odd,
                                 valsM, &codeM, &e2M);

            const bool useP = e2P < e2M; // sqrt is monotone -> compare squared errs
#pragma unroll
            for (int i = 0; i < 8; ++i)
                out_vals[(size_t)r * 8 + i] = useP ? (valsP[i] - 0.25f) : (valsM[i] + 0.25f);
            out_codes[r] = useP ? codeP : codeM;
        }
    }
}

extern "C" void kernel_launch(void* const* d_in, const int* in_sizes, int n_in,
                              void* d_out, int out_size, void* d_ws, size_t ws_size,
                              hipStream_t stream)
{
    const float*         x     = (const float*)d_in[0];         // [N,8]
    const float*         grid  = (const float*)d_in[1];         // [G,8]
    const float*         gnorm = (const float*)d_in[2];         // [G]
    const int*           pam   = (const int*)d_in[3];           // [G]
    const unsigned char* gao   = (const unsigned char*)d_in[4]; // [256] bool
    // d_in[5] = bit_map, folded into the kernel as constants.

    const int N = in_sizes[0] / 8;
    const int G = in_sizes[1] / 8;

    float* out_vals  = (float*)d_out;                  // N*8 floats
    int*   out_codes = (int*)d_out + (size_t)N * 8;    // N int32 codes (bit-placed)

    const int blocks = (N + ROWS_PER_BLOCK - 1) / ROWS_PER_BLOCK;
    hipLaunchKernelGGL(e8p12_quantize_wmma_kernel, dim3(blocks), dim3(256), 0, stream,
                       x, grid, gnorm, pam, gao, out_vals, out_codes, N, G);
}